// NWJ_35046933135407
// MI455X (gfx1250) — compile-verified
//
#include <hip/hip_runtime.h>

#define NS   1024     // N samples
#define XD   256      // X_DIM == Y_DIM
#define HH   256      // HIDDEN
#define ITILE 32      // pairwise i rows per block
#define JTILE 64      // pairwise j cols per block
#define NJT  (NS / JTILE)   // 16 j-tiles (partial-sum slots)

typedef __attribute__((ext_vector_type(2))) float v2f;
typedef __attribute__((ext_vector_type(8))) float v8f;

// ---------------------------------------------------------------------------
// Kernel 1: projection GEMMs via V_WMMA_F32_16X16X4_F32 (full f32 matrix core)
//   z==0:  PXB = x_samples @ W1[:256]  + b1      [NS][HH]
//   z==1:  PY  = y_samples @ W1[256:]            [NS][HH]
// grid (HH/16, NS/16, 2), block = 32 (one wave per 16x16 output tile).
// A fragment (16x4 f32): lane l -> M = l&15, K = k0 + (l>>4)*2 + {0,1}
// B fragment (4x16 f32): lane l -> N = l&15, K = k0 + (l>>4)*2 + {0,1}
// C/D (16x16 f32, 8 VGPRs): lanes 0-15: M=v, N=lane; lanes 16-31: M=v+8.
// ---------------------------------------------------------------------------
__global__ void proj_gemm(const float* __restrict__ X,
                          const float* __restrict__ Y,
                          const float* __restrict__ W1,   // [2*XD][HH] row-major
                          const float* __restrict__ b1,   // [HH]
                          float* __restrict__ PXB,
                          float* __restrict__ PY)
{
    const int lane = threadIdx.x;            // 0..31, one wave
    const int n0 = blockIdx.x * 16;
    const int m0 = blockIdx.y * 16;
    const int zz = blockIdx.z;

    const float* A = (zz == 0) ? X : Y;
    const float* W = W1 + (size_t)zz * XD * HH;

    // Stage the 16xXD A-tile in LDS (coalesced float4 global loads).
    __shared__ __align__(16) float sA[16][XD + 2];   // +2 keeps b64 reads spread
    for (int t = lane; t < 16 * (XD / 4); t += 32) {
        const int r = t / (XD / 4);
        const int c = (t % (XD / 4)) * 4;
        const float4 v = *reinterpret_cast<const float4*>(A + (size_t)(m0 + r) * XD + c);
        sA[r][c + 0] = v.x; sA[r][c + 1] = v.y;
        sA[r][c + 2] = v.z; sA[r][c + 3] = v.w;
    }
    __syncthreads();

    const int l15   = lane & 15;
    const int khalf = (lane >> 4) * 2;       // 0 or 2

    v8f c = {};
    for (int k = 0; k < XD; k += 4) {
        const int kk = k + khalf;
        v2f a, b;
        a.x = sA[l15][kk];
        a.y = sA[l15][kk + 1];
        b.x = W[(size_t)(kk + 0) * HH + n0 + l15];   // coalesced across lanes 0-15
        b.y = W[(size_t)(kk + 1) * HH + n0 + l15];
        c = __builtin_amdgcn_wmma_f32_16x16x4_f32(false, a, false, b,
                                                  (short)0, c, false, false);
    }

    const int mbase = (lane >> 4) * 8;
    const float bias = (zz == 0) ? b1[n0 + l15] : 0.0f;
    float* OUT = (zz == 0) ? PXB : PY;
    #pragma unroll
    for (int v = 0; v < 8; ++v) {
        OUT[(size_t)(m0 + mbase + v) * HH + n0 + l15] = c[v] + bias;
    }
}

// ---------------------------------------------------------------------------
// Kernel 2: pairwise term t1[i,j] = sum_h relu(PY[i,h]+PXB[j,h])*w2[h] + (b2-1)
// Emits per-(i, j-tile) partial sums of exp(t1) (deterministic, no atomics)
// and the diagonal t1[i,i].
// grid (NJT, NS/ITILE), block 256.  tx = j-quad (16), ty = i (16, +16 pair).
// LDS padding chosen for conflict-free reads (see analysis).
// ---------------------------------------------------------------------------
__global__ void pairwise(const float* __restrict__ PY,
                         const float* __restrict__ PXB,
                         const float* __restrict__ W2,   // [HH]
                         const float* __restrict__ b2,   // [1]
                         float* __restrict__ partial,    // [NS][NJT]
                         float* __restrict__ diag)       // [NS]
{
    __shared__ __align__(16) float sPY[ITILE][HH + 4];   // float4 reads, spread banks
    __shared__ __align__(16) float sPXB[JTILE][HH + 1];  // scalar reads, spread banks
    __shared__ __align__(16) float sW2[HH];

    const int tid = threadIdx.x;
    const int tx = tid & 15;
    const int ty = tid >> 4;
    const int i0 = blockIdx.y * ITILE;
    const int j0 = blockIdx.x * JTILE;

    for (int t = tid; t < ITILE * (HH / 4); t += 256) {
        const int r = t / (HH / 4), c = (t % (HH / 4)) * 4;
        *reinterpret_cast<float4*>(&sPY[r][c]) =
            *reinterpret_cast<const float4*>(PY + (size_t)(i0 + r) * HH + c);
    }
    for (int t = tid; t < JTILE * (HH / 4); t += 256) {
        const int r = t / (HH / 4), c = (t % (HH / 4)) * 4;
        const float4 v = *reinterpret_cast<const float4*>(PXB + (size_t)(j0 + r) * HH + c);
        sPXB[r][c + 0] = v.x; sPXB[r][c + 1] = v.y;
        sPXB[r][c + 2] = v.z; sPXB[r][c + 3] = v.w;
    }
    for (int t = tid; t < HH; t += 256) sW2[t] = W2[t];
    __syncthreads();

    const int ia = i0 + ty;        // this thread's two i rows
    const int ib = ia + 16;
    const int jj = tx * 4;         // local j base (4 j's per thread)

    float acc[2][4] = {};
    for (int h = 0; h < HH; h += 4) {
        const float4 w  = *reinterpret_cast<const float4*>(&sW2[h]);
        const float4 pa = *reinterpret_cast<const float4*>(&sPY[ty][h]);
        const float4 pb = *reinterpret_cast<const float4*>(&sPY[ty + 16][h]);
        #pragma unroll
        for (int q = 0; q < 4; ++q) {
            const float x0 = sPXB[jj + q][h + 0];
            const float x1 = sPXB[jj + q][h + 1];
            const float x2 = sPXB[jj + q][h + 2];
            const float x3 = sPXB[jj + q][h + 3];
            acc[0][q] += fmaxf(pa.x + x0, 0.0f) * w.x
                       + fmaxf(pa.y + x1, 0.0f) * w.y
                       + fmaxf(pa.z + x2, 0.0f) * w.z
                       + fmaxf(pa.w + x3, 0.0f) * w.w;
            acc[1][q] += fmaxf(pb.x + x0, 0.0f) * w.x
                       + fmaxf(pb.y + x1, 0.0f) * w.y
                       + fmaxf(pb.z + x2, 0.0f) * w.z
                       + fmaxf(pb.w + x3, 0.0f) * w.w;
        }
    }

    const float c1 = b2[0] - 1.0f;
    float e0 = 0.0f, e1 = 0.0f;
    #pragma unroll
    for (int q = 0; q < 4; ++q) {
        const float t1a = acc[0][q] + c1;
        const float t1b = acc[1][q] + c1;
        e0 += __expf(t1a);
        e1 += __expf(t1b);
        const int j = j0 + jj + q;
        if (j == ia) diag[ia] = t1a;   // t0_i = t1[i,i] + 1
        if (j == ib) diag[ib] = t1b;
    }

    // Reduce over the 16 tx lanes (stays inside each 16-lane half of the wave).
    #pragma unroll
    for (int m = 1; m < 16; m <<= 1) {
        e0 += __shfl_xor(e0, m, 32);
        e1 += __shfl_xor(e1, m, 32);
    }
    if (tx == 0) {
        partial[(size_t)ia * NJT + blockIdx.x] = e0;
        partial[(size_t)ib * NJT + blockIdx.x] = e1;
    }
}

// ---------------------------------------------------------------------------
// Kernel 3: final reduction.
// lower = mean(diag) + 1 - (sum of all exp partials) / N^2
// ---------------------------------------------------------------------------
__global__ void finalize(const float* __restrict__ diag,
                         const float* __restrict__ partial,
                         float* __restrict__ out)
{
    __shared__ float sd[256];
    __shared__ float se[256];
    const int t = threadIdx.x;
    float d = 0.0f, s = 0.0f;
    for (int i = t; i < NS; i += 256)        d += diag[i];
    for (int i = t; i < NS * NJT; i += 256)  s += partial[i];
    sd[t] = d; se[t] = s;
    __syncthreads();
    for (int k = 128; k > 0; k >>= 1) {
        if (t < k) { sd[t] += sd[t + k]; se[t] += se[t + k]; }
        __syncthreads();
    }
    if (t == 0) {
        const float nn = (float)NS;
        out[0] = sd[0] / nn + 1.0f - se[0] / (nn * nn);
    }
}

// ---------------------------------------------------------------------------
extern "C" void kernel_launch(void* const* d_in, const int* in_sizes, int n_in,
                              void* d_out, int out_size, void* d_ws, size_t ws_size,
                              hipStream_t stream)
{
    const float* X  = (const float*)d_in[0];   // [NS, XD]
    const float* Y  = (const float*)d_in[1];   // [NS, XD]
    const float* W1 = (const float*)d_in[2];   // [2*XD, HH]
    const float* b1 = (const float*)d_in[3];   // [HH]
    const float* W2 = (const float*)d_in[4];   // [HH, 1]
    const float* b2 = (const float*)d_in[5];   // [1]
    float* out = (float*)d_out;

    // Workspace layout (all fully overwritten each call; ~2.1 MB)
    float* PXB     = (float*)d_ws;                          // NS*HH
    float* PY      = PXB + (size_t)NS * HH;                 // NS*HH
    float* partial = PY  + (size_t)NS * HH;                 // NS*NJT
    float* diag    = partial + (size_t)NS * NJT;            // NS

    proj_gemm<<<dim3(HH / 16, NS / 16, 2), 32, 0, stream>>>(X, Y, W1, b1, PXB, PY);
    pairwise<<<dim3(NJT, NS / ITILE), 256, 0, stream>>>(PY, PXB, W2, b2, partial, diag);
    finalize<<<1, 256, 0, stream>>>(diag, partial, out);
}